// RNN_41626823033250
// MI455X (gfx1250) — compile-verified
//
#include <hip/hip_runtime.h>

typedef __attribute__((ext_vector_type(2))) float v2f;
typedef __attribute__((ext_vector_type(4))) float v4f;
typedef __attribute__((ext_vector_type(8))) float v8f;

#ifndef __has_builtin
#define __has_builtin(x) 0
#endif

// ---------------------------------------------------------------------------
// Row (M) permutation of the 28 gate rows across two 16-row WMMA tiles.
// After D = A*B + C (M=gate, N=batch; C/D layout: vgpr j holds row j in lanes
// 0-15 and row j+8 in lanes 16-31):
//   I_s = D0[s]   : low lanes -> i_s,   high lanes -> i_{4+s}
//   F_s = D0[4+s] : low lanes -> f_s,   high lanes -> f_{4+s}
//   G_s = D1[s]   : low lanes -> g_s,   high lanes -> g_{4+s}
//   O_s = D1[4+s] : low lanes -> o_s,   high lanes -> o_{4+s}
// Lane b computes hidden units 0..3 of batch b, lane b+16 computes units 4..6
// (slot 3 high is a zero-padded dummy).  No cross-lane traffic per timestep.
// Original gate row order (PyTorch): i=0..6, f=7..13, g=14..20, o=21..27.
__constant__ signed char GMAP[32] = {
    0,  1,  2,  3,   // m0..3   : i0..i3
    7,  8,  9,  10,  // m4..7   : f0..f3
    4,  5,  6,  -1,  // m8..11  : i4..i6, pad
    11, 12, 13, -1,  // m12..15 : f4..f6, pad
    14, 15, 16, 17,  // m16..19 : g0..g3
    21, 22, 23, 24,  // m20..23 : o0..o3
    18, 19, 20, -1,  // m24..27 : g4..g6, pad
    25, 26, 27, -1   // m28..31 : o4..o6, pad
};
// Column (K) permutation of the augmented input [h0..h6, x], chosen so the B
// matrix (4x16 per K-chunk; vgpr v: low lanes = row 4c+v, high = row 4c+v+2)
// is exactly the live h registers (low lanes keep h0..h3, high lanes h4..h6):
//   K: 0=h0 1=h1 2=h4 3=h5 4=h2 5=h3 6=h6 7=x
__constant__ signed char KMAP[8] = {0, 1, 4, 5, 2, 3, 6, 7};  // 7 == x column

__device__ __forceinline__ float dev_tanh(float x) {
#if __has_builtin(__builtin_amdgcn_tanhf)
  return __builtin_amdgcn_tanhf(x);  // gfx1250 v_tanh_f32 (trans pipe)
#else
  float e = __builtin_amdgcn_exp2f(x * 2.88539008177792681472f);  // exp(2x)
  return 1.0f - 2.0f * __builtin_amdgcn_rcpf(e + 1.0f);
#endif
}

#define T_LEN 2048
#define H_DIM 7

__global__ __launch_bounds__(256) void lstm_wmma_kernel(
    const float* __restrict__ x,      // [16384, 2048, 1]
    const float* __restrict__ W_ih,   // [28, 1]
    const float* __restrict__ W_hh,   // [28, 7]
    const float* __restrict__ b_ih,   // [28]
    const float* __restrict__ b_hh,   // [28]
    const float* __restrict__ W_fc,   // [1, 7]
    const float* __restrict__ b_fc,   // [1]
    float* __restrict__ out)          // [16384]
{
  const int lane  = threadIdx.x & 31;
  const int wave  = threadIdx.x >> 5;
  const int bBase = (blockIdx.x * 8 + wave) * 16;  // 16 batch rows per wave
  const bool hi   = lane >= 16;
  const int  mrow = lane & 15;

  // ---- Build constant A tiles (permuted [W_hh | W_ih], 32x8 over 2 tiles).
  // Sigmoid gates (i/f/o) are pre-scaled by 0.5 so the per-step sigmoid is
  // just fma(tanh(v), 0.5, 0.5); tanh gates (g, rows 14..20) keep full scale.
  // A 16x4 f32 layout: lane l holds row M=l%16; vgpr0 = K=4c+(l<16?0:2),
  // vgpr1 = K+1.
  v2f A[2][2];
#pragma unroll
  for (int t = 0; t < 2; ++t) {
#pragma unroll
    for (int c = 0; c < 2; ++c) {
      float e[2];
#pragma unroll
      for (int p = 0; p < 2; ++p) {
        int m = mrow + 16 * t;
        int g = GMAP[m];
        int k = 4 * c + (hi ? 2 : 0) + p;
        float v = 0.0f;
        if (g >= 0) {
          int   kk = KMAP[k];
          float sc = (g >= 14 && g <= 20) ? 1.0f : 0.5f;
          v = sc * ((kk == 7) ? W_ih[g] : W_hh[g * H_DIM + kk]);
        }
        e[p] = v;
      }
      A[t][c] = (v2f){e[0], e[1]};
    }
  }

  // ---- Bias accumulator tiles (C layout: vgpr j = row j / j+8), same scale.
  v8f CB[2];
#pragma unroll
  for (int t = 0; t < 2; ++t) {
#pragma unroll
    for (int j = 0; j < 8; ++j) {
      int m = j + (hi ? 8 : 0) + 16 * t;
      int g = GMAP[m];
      float v = 0.0f;
      if (g >= 0) {
        float sc = (g >= 14 && g <= 20) ? 1.0f : 0.5f;
        v = sc * (b_ih[g] + b_hh[g]);
      }
      CB[t][j] = v;
    }
  }

  // ---- LSTM state: low lanes units 0..3, high lanes units 4..6 (+dummy) ---
  float hS[4] = {0.0f, 0.0f, 0.0f, 0.0f};
  float cS[4] = {0.0f, 0.0f, 0.0f, 0.0f};

  // Every lane loads the full 8-step x chunk of its batch (lane pair loads
  // identical addresses -> coalesced; no cross-lane exchange needed).
  // Non-temporal (streamed once), software-pipelined TWO chunks ahead so the
  // load->use distance stays >= one full 8-step body even after scheduling.
  const float* xrow = x + (size_t)(bBase + mrow) * T_LEN;
  v4f q0a = __builtin_nontemporal_load((const v4f*)(xrow + 0));
  v4f q0b = __builtin_nontemporal_load((const v4f*)(xrow + 4));
  v4f q1a = __builtin_nontemporal_load((const v4f*)(xrow + 8));
  v4f q1b = __builtin_nontemporal_load((const v4f*)(xrow + 12));

  for (int tc = 0; tc < T_LEN; tc += 8) {
    int tp = tc + 16;
    tp = (tp < T_LEN) ? tp : (T_LEN - 8);  // tail: re-load last chunk
    v4f na = __builtin_nontemporal_load((const v4f*)(xrow + tp));
    v4f nb = __builtin_nontemporal_load((const v4f*)(xrow + tp + 4));

    const v4f xa = q0a;
    const v4f xb = q0b;

#pragma unroll
    for (int j = 0; j < 8; ++j) {
      const float xj = (j < 4) ? xa[j] : xb[j - 4];  // x[b][tc+j], all lanes

      // B chunks fall out of the live registers by construction (KMAP).
      v2f B0 = (v2f){hS[0], hS[1]};              // K rows 0..3
      v2f B1 = (v2f){hS[2], hi ? xj : hS[3]};    // K rows 4..7 (7 = x)

      // gates = Wperm * [h;x] + bias  (28 gates x 16 batches, f32 WMMA)
      v8f D0 = __builtin_amdgcn_wmma_f32_16x16x4_f32(
          false, A[0][0], false, B0, (short)0, CB[0], false, false);
      D0 = __builtin_amdgcn_wmma_f32_16x16x4_f32(
          false, A[0][1], false, B1, (short)0, D0, false, false);
      v8f D1 = __builtin_amdgcn_wmma_f32_16x16x4_f32(
          false, A[1][0], false, B0, (short)0, CB[1], false, false);
      v8f D1b = __builtin_amdgcn_wmma_f32_16x16x4_f32(
          false, A[1][1], false, B1, (short)0, D1, false, false);

      // Elementwise LSTM update, 4 unit-slots per lane, no shuffles.
      // i/f/o rows were pre-scaled by 0.5: sigmoid = 0.5*tanh(v) + 0.5.
#pragma unroll
      for (int s = 0; s < 4; ++s) {
        float si = 0.5f * dev_tanh(D0[s]) + 0.5f;
        float sf = 0.5f * dev_tanh(D0[4 + s]) + 0.5f;
        float tg = dev_tanh(D1b[s]);
        float so = 0.5f * dev_tanh(D1b[4 + s]) + 0.5f;
        float cn = sf * cS[s] + si * tg;
        cS[s]    = cn;
        hS[s]    = so * dev_tanh(cn);
      }
    }

    // rotate the 2-deep prefetch queue
    q0a = q1a; q0b = q1b;
    q1a = na;  q1b = nb;
  }

  // ---- Head: out[b] = W_fc . h + b_fc (pair reduction over lane halves) ---
  float wf[4];
#pragma unroll
  for (int s = 0; s < 4; ++s) {
    int idx = s + (hi ? 4 : 0);
    wf[s] = (idx < H_DIM) ? W_fc[idx] : 0.0f;  // masks the dummy unit
  }
  float p = hS[0] * wf[0] + hS[1] * wf[1] + hS[2] * wf[2] + hS[3] * wf[3];
  p += __shfl_xor(p, 16, 32);
  if (!hi) out[bBase + mrow] = p + b_fc[0];
}

extern "C" void kernel_launch(void* const* d_in, const int* in_sizes, int n_in,
                              void* d_out, int out_size, void* d_ws,
                              size_t ws_size, hipStream_t stream) {
  (void)in_sizes; (void)n_in; (void)out_size; (void)d_ws; (void)ws_size;
  const float* x    = (const float*)d_in[0];
  const float* W_ih = (const float*)d_in[1];
  const float* W_hh = (const float*)d_in[2];
  const float* b_ih = (const float*)d_in[3];
  const float* b_hh = (const float*)d_in[4];
  const float* W_fc = (const float*)d_in[5];
  const float* b_fc = (const float*)d_in[6];
  float* out = (float*)d_out;

  // 16384 batches / (16 per wave * 8 waves per block) = 128 blocks
  lstm_wmma_kernel<<<dim3(128), dim3(256), 0, stream>>>(
      x, W_ih, W_hh, b_ih, b_hh, W_fc, b_fc, out);
}